// SubwordStackLSTMCell_48455821033604
// MI455X (gfx1250) — compile-verified
//
#include <hip/hip_runtime.h>

#define B_ 1024
#define S_ 258
#define I_ 512
#define H_ 512

typedef __attribute__((ext_vector_type(2))) float v2f;
typedef __attribute__((ext_vector_type(4))) float f4;
typedef __attribute__((ext_vector_type(8))) float v8f;

// Sub-fragment selection out of a b128 load: components [0,1] -> WMMA step A,
// components [2,3] -> WMMA step B (valid because the K-reduction order inside
// a WMMA is free as long as A and B agree on the k <-> slot mapping).
#define LO2(v) __builtin_shufflevector((v), (v), 0, 1)
#define HI2(v) __builtin_shufflevector((v), (v), 2, 3)
#define WMMA_F32(a, b, c) \
    __builtin_amdgcn_wmma_f32_16x16x4_f32(false, (a), false, (b), (short)0, (c), false, false)

__device__ __forceinline__ float fast_tanh(float x) {
#if __has_builtin(__builtin_amdgcn_tanhf)
    return __builtin_amdgcn_tanhf(x);      // V_TANH_F32 (TRANS pipe)
#else
    return tanhf(x);
#endif
}
__device__ __forceinline__ float fast_sigmoid(float x) {
    return __builtin_amdgcn_rcpf(1.0f + __expf(-x));   // v_exp_f32 + v_rcp_f32
}

// ---------------------------------------------------------------------------
// Kernel 1: bulk stack copy (input stack -> output stack), non-temporal b128.
// 2 x 540 MB moved twice (read+write) => the dominant cost (~93us @ 23.3TB/s).
// ---------------------------------------------------------------------------
__global__ __launch_bounds__(256) void copy_stack_kernel(
    const f4* __restrict__ src, f4* __restrict__ dst, long n4)
{
    long i = (long)blockIdx.x * blockDim.x + threadIdx.x;
    long stride = (long)gridDim.x * blockDim.x;
    for (; i < n4; i += stride) {
        f4 v = __builtin_nontemporal_load(&src[i]);
        __builtin_nontemporal_store(v, &dst[i]);
    }
}

// ---------------------------------------------------------------------------
// Kernel 2: fused dual-LSTM-cell via V_WMMA_F32_16X16X4_F32.
//   gates_r = char @ W_ih_r^T + h_gathered @ W_hh_r^T + b_ih_r + b_hh_r
//   gates_l = char @ W_ih_l^T + b_ih_l + b_hh_l   (h=0 -> W_hh_l dead,
//                                                  left 'f' gate dead)
// Each wave owns a 32(m) x 16(n) slab = two 16x16 tiles sharing B fragments;
// K is consumed 8 at a time via b128 loads (2 WMMA steps per load).
// Epilogue: activations on the TRANS pipe, scatter h_r/c_r at pos+1 into the
// output stacks, write [h_r | h_l] row-major into ws_cat (B x 2H).
// ---------------------------------------------------------------------------
__global__ __launch_bounds__(128) void lstm_wmma_kernel(
    const float* __restrict__ xchar,
    const float* __restrict__ stack_h,
    const float* __restrict__ stack_c,
    const int*   __restrict__ pos,
    const float* __restrict__ W_ih_r, const float* __restrict__ W_hh_r,
    const float* __restrict__ b_ih_r, const float* __restrict__ b_hh_r,
    const float* __restrict__ W_ih_l,
    const float* __restrict__ b_ih_l, const float* __restrict__ b_hh_l,
    float* __restrict__ out_sh, float* __restrict__ out_sc,
    float* __restrict__ ws_cat)
{
    const int lane  = threadIdx.x & 31;
    const int wave  = threadIdx.x >> 5;
    const int tile  = blockIdx.x * 4 + wave;   // 32 m-pairs * 32 ntiles = 1024
    const int mpair = tile >> 5;
    const int ntile = tile & 31;
    const int m0 = mpair * 32, n0 = ntile * 16;
    const int lm   = lane & 15;                // m (for A) / n (for B,C,D)
    const int half = lane >> 4;                // K-half selector

    // Per-lane A row pointers for both m-tiles (h rows gathered via pos).
    const float* ax[2];
    const float* ah[2];
#pragma unroll
    for (int t = 0; t < 2; ++t) {
        const int row = m0 + t * 16 + lm;
        ax[t] = xchar + (size_t)row * I_;
        ah[t] = stack_h + ((size_t)row * S_ + pos[row]) * H_;
    }

    v8f accr[2][4] = {};   // right cell gates i,f,g,o per m-tile
    v8f accl[2][4] = {};   // left cell gates (index 1 unused)

    for (int k0 = 0; k0 < I_; k0 += 8) {
        const int ka = k0 + half * 4;          // this half's 4 consecutive k's
        f4 Ax[2], Ah[2];
#pragma unroll
        for (int t = 0; t < 2; ++t) {
            Ax[t] = *(const f4*)(ax[t] + ka);
            Ah[t] = *(const f4*)(ah[t] + ka);
        }
#pragma unroll
        for (int g = 0; g < 4; ++g) {
            const size_t wrow = (size_t)(g * H_ + n0 + lm);   // W row = gate col n
            const f4 Bi = *(const f4*)(W_ih_r + wrow * I_ + ka);
            const f4 Bh = *(const f4*)(W_hh_r + wrow * H_ + ka);
#pragma unroll
            for (int t = 0; t < 2; ++t) {
                accr[t][g] = WMMA_F32(LO2(Ax[t]), LO2(Bi), accr[t][g]);
                accr[t][g] = WMMA_F32(HI2(Ax[t]), HI2(Bi), accr[t][g]);
                accr[t][g] = WMMA_F32(LO2(Ah[t]), LO2(Bh), accr[t][g]);
                accr[t][g] = WMMA_F32(HI2(Ah[t]), HI2(Bh), accr[t][g]);
            }
            if (g != 1) {                      // left 'f' gate multiplies c=0
                const f4 Bl = *(const f4*)(W_ih_l + wrow * I_ + ka);
#pragma unroll
                for (int t = 0; t < 2; ++t) {
                    accl[t][g] = WMMA_F32(LO2(Ax[t]), LO2(Bl), accl[t][g]);
                    accl[t][g] = WMMA_F32(HI2(Ax[t]), HI2(Bl), accl[t][g]);
                }
            }
        }
    }

    // ---- epilogue ----
    const int n = n0 + lm;
    float br[4], bl[4];
#pragma unroll
    for (int g = 0; g < 4; ++g) {
        br[g] = b_ih_r[g * H_ + n] + b_hh_r[g * H_ + n];
        bl[g] = b_ih_l[g * H_ + n] + b_hh_l[g * H_ + n];
    }

#pragma unroll
    for (int t = 0; t < 2; ++t) {
#pragma unroll
        for (int v = 0; v < 8; ++v) {
            // D layout: VGPR v holds M=v (lanes 0-15) / M=v+8 (lanes 16-31)
            const int m  = m0 + t * 16 + v + half * 8;
            const int pm = pos[m];
            const size_t base_old = ((size_t)m * S_ + pm) * H_ + n;
            const size_t base_new = ((size_t)m * S_ + pm + 1) * H_ + n;

            // right cell (PyTorch gate order i,f,g,o)
            const float gi = accr[t][0][v] + br[0];
            const float gf = accr[t][1][v] + br[1];
            const float gg = accr[t][2][v] + br[2];
            const float go = accr[t][3][v] + br[3];
            const float c_old = stack_c[base_old];
            const float c_new = fast_sigmoid(gf) * c_old + fast_sigmoid(gi) * fast_tanh(gg);
            const float h_new = fast_sigmoid(go) * fast_tanh(c_new);
            out_sc[base_new] = c_new;
            out_sh[base_new] = h_new;
            ws_cat[(size_t)m * (2 * H_) + n] = h_new;

            // left cell with zero state: c_l = sigmoid(i)*tanh(g)
            const float li = accl[t][0][v] + bl[0];
            const float lg = accl[t][2][v] + bl[2];
            const float lo = accl[t][3][v] + bl[3];
            const float cl = fast_sigmoid(li) * fast_tanh(lg);
            const float hl = fast_sigmoid(lo) * fast_tanh(cl);
            ws_cat[(size_t)m * (2 * H_) + H_ + n] = hl;
        }
    }
}

// ---------------------------------------------------------------------------
// Kernel 3: subword = tanh([h_r, h_l] @ W_comp^T + b_comp)
// M=1024, N=512, K=1024 NT GEMM; same 32x16 slab per wave, b128 K-chunks.
// ---------------------------------------------------------------------------
__global__ __launch_bounds__(128) void compose_wmma_kernel(
    const float* __restrict__ cat,
    const float* __restrict__ W_comp,
    const float* __restrict__ b_comp,
    float* __restrict__ out)
{
    const int lane  = threadIdx.x & 31;
    const int wave  = threadIdx.x >> 5;
    const int tile  = blockIdx.x * 4 + wave;   // 32 m-pairs * 32 ntiles = 1024
    const int mpair = tile >> 5;
    const int ntile = tile & 31;
    const int m0 = mpair * 32, n0 = ntile * 16;
    const int lm   = lane & 15;
    const int half = lane >> 4;

    const float* arow0 = cat + (size_t)(m0 + lm) * (2 * H_);
    const float* arow1 = cat + (size_t)(m0 + 16 + lm) * (2 * H_);
    const float* brow  = W_comp + (size_t)(n0 + lm) * (2 * H_);

    v8f acc[2] = {};
    for (int k0 = 0; k0 < 2 * H_; k0 += 8) {
        const int ka = k0 + half * 4;
        const f4 A0 = *(const f4*)(arow0 + ka);
        const f4 A1 = *(const f4*)(arow1 + ka);
        const f4 Bv = *(const f4*)(brow + ka);
        acc[0] = WMMA_F32(LO2(A0), LO2(Bv), acc[0]);
        acc[0] = WMMA_F32(HI2(A0), HI2(Bv), acc[0]);
        acc[1] = WMMA_F32(LO2(A1), LO2(Bv), acc[1]);
        acc[1] = WMMA_F32(HI2(A1), HI2(Bv), acc[1]);
    }

    const int n = n0 + lm;
    const float bb = b_comp[n];
#pragma unroll
    for (int t = 0; t < 2; ++t) {
#pragma unroll
        for (int v = 0; v < 8; ++v) {
            const int m = m0 + t * 16 + v + half * 8;
            out[(size_t)m * H_ + n] = fast_tanh(acc[t][v] + bb);
        }
    }
}

// ---------------------------------------------------------------------------
extern "C" void kernel_launch(void* const* d_in, const int* in_sizes, int n_in,
                              void* d_out, int out_size, void* d_ws, size_t ws_size,
                              hipStream_t stream) {
    const float* xchar   = (const float*)d_in[0];
    const float* stack_h = (const float*)d_in[1];
    const float* stack_c = (const float*)d_in[2];
    const int*   pos     = (const int*)  d_in[3];
    const float* W_ih_r  = (const float*)d_in[4];
    const float* W_hh_r  = (const float*)d_in[5];
    const float* b_ih_r  = (const float*)d_in[6];
    const float* b_hh_r  = (const float*)d_in[7];
    const float* W_ih_l  = (const float*)d_in[8];
    // d_in[9] = W_hh_l: multiplied by zero state, unused.
    const float* b_ih_l  = (const float*)d_in[10];
    const float* b_hh_l  = (const float*)d_in[11];
    const float* W_comp  = (const float*)d_in[12];
    const float* b_comp  = (const float*)d_in[13];

    float* out_sub = (float*)d_out;                       // B*H
    float* out_sh  = out_sub + (size_t)B_ * H_;           // B*S*H
    float* out_sc  = out_sh  + (size_t)B_ * S_ * H_;      // B*S*H
    float* ws_cat  = (float*)d_ws;                        // B * 2H floats (4 MB)

    const long n4 = (long)B_ * S_ * H_ / 4;
    copy_stack_kernel<<<4096, 256, 0, stream>>>((const f4*)stack_h, (f4*)out_sh, n4);
    copy_stack_kernel<<<4096, 256, 0, stream>>>((const f4*)stack_c, (f4*)out_sc, n4);

    // 1024 wave-slabs (32 rows x 16 cols each), 4 waves per block
    lstm_wmma_kernel<<<256, 128, 0, stream>>>(
        xchar, stack_h, stack_c, pos,
        W_ih_r, W_hh_r, b_ih_r, b_hh_r,
        W_ih_l, b_ih_l, b_hh_l,
        out_sh, out_sc, ws_cat);

    compose_wmma_kernel<<<256, 128, 0, stream>>>(ws_cat, W_comp, b_comp, out_sub);
}